// ConcatAttention_30777735643475
// MI455X (gfx1250) — compile-verified
//
#include <hip/hip_runtime.h>
#include <hip/hip_bf16.h>
#include <stddef.h>
#include <stdint.h>

// Problem constants (B, L, H, E) = (2, 2048, 16, 64)
constexpr int Bc = 2;
constexpr int Lc = 2048;
constexpr int Hc = 16;
constexpr int Ec = 64;
constexpr int NCHUNK = 16;   // chunks per (b,h) sequence
constexpr int CHS    = 128;  // rows per chunk
constexpr int HE     = Hc * Ec;

typedef float v2f __attribute__((ext_vector_type(2)));
typedef float v8f __attribute__((ext_vector_type(8)));

// ---------------------------------------------------------------------------
// Kernel A: per (b,h): a_k[s] = K[b,s,h,:] . w_k ; m = max_s a_k ;
//           e[s] = exp(scale*(a_k[s]-m))  -> ws_e[bh*L + s]
// ---------------------------------------------------------------------------
__global__ void ka_escore(const float* __restrict__ K,
                          const float* __restrict__ w_score,
                          float* __restrict__ ws_e) {
    const int bh = blockIdx.x;            // 0..31
    const int b = bh / Hc, h = bh % Hc;
    const int t = threadIdx.x;            // 256 threads

    __shared__ float a_lds[Lc];
    __shared__ float red[256];
    __shared__ float wk[Ec];

    if (t < Ec) wk[t] = w_score[Ec + t];  // w_k = w_score[E:2E]
    __syncthreads();

    float lmax = -3.4e38f;
    for (int s = t; s < Lc; s += 256) {
        const float* kp = K + (((size_t)b * Lc + s) * Hc + h) * Ec;
        float acc = 0.f;
#pragma unroll
        for (int e = 0; e < Ec; e += 4) {
            float4 kv = *(const float4*)(kp + e);
            acc += kv.x * wk[e] + kv.y * wk[e + 1] + kv.z * wk[e + 2] + kv.w * wk[e + 3];
        }
        a_lds[s] = acc;
        lmax = fmaxf(lmax, acc);
    }
    red[t] = lmax;
    __syncthreads();
    for (int off = 128; off > 0; off >>= 1) {
        if (t < off) red[t] = fmaxf(red[t], red[t + off]);
        __syncthreads();
    }
    const float m = red[0];
    const float scale = 0.125f;           // 1/sqrt(64)
    for (int s = t; s < Lc; s += 256) {
        ws_e[(size_t)bh * Lc + s] = __expf(scale * (a_lds[s] - m));
    }
}

// ---------------------------------------------------------------------------
// Kernel B: per chunk: chunkV[d] = sum_{s in chunk} e[s]*V[b,s,h,d]  (64 floats)
//           chunkE    = sum_{s in chunk} e[s]
// ---------------------------------------------------------------------------
__global__ void kb_chunksum(const float* __restrict__ V,
                            const float* __restrict__ ws_e,
                            float* __restrict__ ws_cv,
                            float* __restrict__ ws_ce) {
    const int blk = blockIdx.x;           // 0..511
    const int bh = blk / NCHUNK;
    const int c  = blk % NCHUNK;
    const int b = bh / Hc, h = bh % Hc;
    const int t = threadIdx.x;            // 256 threads
    const int d = t & 63, g = t >> 6;     // 64 d-lanes x 4 s-groups
    const int base = c * CHS;

    __shared__ float part[256];
    __shared__ float es[CHS];

    if (t < CHS) es[t] = ws_e[(size_t)bh * Lc + base + t];
    __syncthreads();

    float acc = 0.f;
    for (int j = 0; j < 32; ++j) {
        const int s = 32 * g + j;
        acc += es[s] * V[(((size_t)b * Lc + base + s) * Hc + h) * Ec + d];
    }
    part[t] = acc;
    __syncthreads();
    if (t < 64) {
        ws_cv[(size_t)blk * Ec + t] = part[t] + part[t + 64] + part[t + 128] + part[t + 192];
    }
    __syncthreads();
    if (t < 64) part[t] = es[t] + es[t + 64];
    __syncthreads();
    for (int off = 32; off > 0; off >>= 1) {
        if (t < off) part[t] += part[t + off];
        __syncthreads();
    }
    if (t == 0) ws_ce[blk] = part[0];
}

// ---------------------------------------------------------------------------
// Kernel D: per chunk: carry-in from previous chunks, then 8 sequential 16-row
// sub-tiles. intra(m,d) = sum_{s<=m} e_s*V(s,d) via V_WMMA_F32_16X16X4_F32
// with A[m][s] = (s<=m)? e_s : 0 (4 K-slices). One wave per 16-wide N-tile.
// The chunk's 128x64 f32 V tile is staged global->LDS with the CDNA5 async
// copy engine (ASYNCcnt), so the WMMA loop runs entirely out of LDS.
// ---------------------------------------------------------------------------
__global__ void kd_output(const float* __restrict__ V,
                          const float* __restrict__ ws_e,
                          const float* __restrict__ ws_cv,
                          const float* __restrict__ ws_ce,
                          float* __restrict__ out) {
    const int blk = blockIdx.x;           // 0..511
    const int bh = blk / NCHUNK;
    const int c  = blk % NCHUNK;
    const int b = bh / Hc, h = bh % Hc;
    const int t = threadIdx.x;            // 128 threads = 4 waves (wave32)
    const int lane = t & 31, wave = t >> 5;
    const int base = c * CHS;

    __shared__ __align__(16) float v_tile[CHS * Ec];   // 32 KB chunk V tile
    __shared__ __align__(16) float e_lds[CHS];
    __shared__ float carryV[Ec];
    __shared__ float carryE_s;

    // ---- async global->LDS staging of the chunk's V tile -------------------
    // 128 threads x 16B = 2KB (8 rows) per issue; 16 issues cover 128 rows.
    {
        const int rowInGrp = t >> 4;              // 0..7
        const int col = (t & 15) * 4;             // float column 0,4,...,60
#pragma unroll
        for (int it = 0; it < 16; ++it) {
            const int s = it * 8 + rowInGrp;
            const uint64_t gaddr = (uint64_t)(uintptr_t)
                (V + (((size_t)b * Lc + base + s) * Hc + h) * Ec + col);
            const uint32_t laddr = (uint32_t)(uintptr_t)&v_tile[s * Ec + col];
            asm volatile("global_load_async_to_lds_b128 %0, %1, off"
                         :: "v"(laddr), "v"(gaddr) : "memory");
        }
    }

    if (t < CHS) e_lds[t] = ws_e[(size_t)bh * Lc + base + t];
    if (t < Ec) {
        float acc = 0.f;
        for (int cc = 0; cc < c; ++cc)
            acc += ws_cv[((size_t)bh * NCHUNK + cc) * Ec + t];
        carryV[t] = acc;
    }
    if (t == 127) {
        float acc = 0.f;
        for (int cc = 0; cc < c; ++cc) acc += ws_ce[bh * NCHUNK + cc];
        carryE_s = acc;
    }
    asm volatile("s_wait_asynccnt 0" ::: "memory");
    __syncthreads();

    // WMMA f32 16x16x4 lane layout (ISA 7.12.2):
    //   A 16x4 : lane -> M = lane&15 ; VGPR0/1 hold K = koff, koff+1 ; koff = (lane<16)?0:2
    //   B 4x16 : lane -> N = lane&15 ; VGPR0/1 hold K = koff, koff+1
    //   C/D    : VGPR r -> M = r + 8*(lane>=16), N = lane&15
    const int half = lane >> 4;
    const int koff = half * 2;
    const int nloc = lane & 15;
    const int d = wave * 16 + nloc;       // this wave's N-tile column
    const int m = lane & 15;              // A row for this lane

    // Loop-invariant causal masks for the 4 K-slices (avoids exec-guarded
    // LDS loads in the hot loop: unconditional b64 load * 0/1 mask instead).
    v2f amask[4];
#pragma unroll
    for (int k = 0; k < 4; ++k) {
        const int s0 = 4 * k + koff;
        amask[k].x = (s0     <= m) ? 1.f : 0.f;
        amask[k].y = (s0 + 1 <= m) ? 1.f : 0.f;
    }

    for (int sub = 0; sub < 8; ++sub) {
        const int esub = sub * 16;
        const float carry = carryV[d];
        const float cE = carryE_s;

        float sum8 = 0.f;
#pragma unroll
        for (int s = 0; s < 8; ++s) sum8 += e_lds[esub + s];
        float acc = half ? sum8 : 0.f;    // exclusive prefix start for this lane-half

        v8f cfrag = {};
#pragma unroll
        for (int k = 0; k < 4; ++k) {
            const int s0 = 4 * k + koff;
            const v2f ep = *(const v2f*)&e_lds[esub + s0];   // 8B-aligned pair
            const v2f afrag = ep * amask[k];
            v2f bfrag;
            bfrag.x = v_tile[(esub + s0) * Ec + d];
            bfrag.y = v_tile[(esub + s0 + 1) * Ec + d];
            cfrag = __builtin_amdgcn_wmma_f32_16x16x4_f32(
                false, afrag, false, bfrag, (short)0, cfrag, false, false);
        }

        float* obase = out + (((size_t)b * Lc + base + esub) * Hc + h) * Ec + d;
#pragma unroll
        for (int r = 0; r < 8; ++r) {
            const int M = 8 * half + r;   // row index of cfrag[r] for this lane-half
            acc += e_lds[esub + M];       // inclusive prefix of e up to row M
            // Denominator is a sum of exponentials: strictly positive, in
            // (0, 2048] -> hardware v_rcp_f32 (~1 ulp) instead of the
            // ~8-op IEEE division expansion per row.
            const float rinv = __builtin_amdgcn_rcpf(cE + acc);
            obase[(size_t)M * HE] = (carry + cfrag[r]) * rinv;
        }

        __syncthreads();
        if (half) carryV[d] = carry + cfrag[7];   // cfrag[7] @ half==1 is row M=15 (full tile sum)
        if (t == 31) carryE_s = cE + acc;         // wave0 lane31: acc == sum of all 16 e
        __syncthreads();
    }
}

// ---------------------------------------------------------------------------
// Launcher. Inputs: 0=queries (unused: cancels in softmax), 1=keys, 2=values,
// 3=w_score (only w_k half used), 4=b_score (unused: cancels).
// Workspace: ws_e (B*H*L) | ws_cv (512*64) | ws_ce (512)  floats.
// ---------------------------------------------------------------------------
extern "C" void kernel_launch(void* const* d_in, const int* in_sizes, int n_in,
                              void* d_out, int out_size, void* d_ws, size_t ws_size,
                              hipStream_t stream) {
    (void)in_sizes; (void)n_in; (void)out_size; (void)ws_size;
    const float* K = (const float*)d_in[1];
    const float* V = (const float*)d_in[2];
    const float* w_score = (const float*)d_in[3];
    float* out = (float*)d_out;

    float* ws_e  = (float*)d_ws;                          // 65536 floats
    float* ws_cv = ws_e + (size_t)Bc * Hc * Lc;           // 32768 floats
    float* ws_ce = ws_cv + (size_t)Bc * Hc * NCHUNK * Ec; // 512 floats

    ka_escore<<<Bc * Hc, 256, 0, stream>>>(K, w_score, ws_e);
    kb_chunksum<<<Bc * Hc * NCHUNK, 256, 0, stream>>>(V, ws_e, ws_cv, ws_ce);
    kd_output<<<Bc * Hc * NCHUNK, 128, 0, stream>>>(V, ws_e, ws_cv, ws_ce, out);
}